// DVBundle_5772436046201
// MI455X (gfx1250) — compile-verified
//
#include <hip/hip_runtime.h>
#include <math.h>
#include <stdint.h>

typedef __attribute__((ext_vector_type(2))) float v2f;
typedef __attribute__((ext_vector_type(4))) float v4f;
typedef __attribute__((ext_vector_type(8))) float v8f;
typedef int v4i_b128 __attribute__((vector_size(16)));   // builtin's b128 payload type

#define N_NEURONS 4096
#define N_INPUTS  8192
#define ROWS      16          // neuron rows per block (WMMA M)
#define CH        128         // i-values staged per LDS chunk (double-buffered)
#define NCHUNK    (N_INPUTS / CH)
#define THREADS   256
#define WAVES     8
#define IPW       (CH / WAVES)
#define ROWPITCH  68          // 16 rows * 4 k + 4 pad floats (conflict-free A reads)

// ---- gfx1250 async global->LDS copy (ASYNCcnt-tracked), with asm fallback ----
__device__ __forceinline__ void async_ld_b128(void* lds_dst, const void* glb_src) {
#if __has_builtin(__builtin_amdgcn_global_load_async_to_lds_b128)
    __builtin_amdgcn_global_load_async_to_lds_b128(
        (__attribute__((address_space(1))) v4i_b128*)(uintptr_t)glb_src,
        (__attribute__((address_space(3))) v4i_b128*)lds_dst, 0, 0);
#else
    asm volatile("global_load_async_to_lds_b128 %0, %1, off"
                 :: "v"((uint32_t)(uintptr_t)(__attribute__((address_space(3))) void*)lds_dst),
                    "v"((uint64_t)(uintptr_t)glb_src)
                 : "memory");
#endif
}

__device__ __forceinline__ void wait_async0() {
#if __has_builtin(__builtin_amdgcn_s_wait_asynccnt)
    __builtin_amdgcn_s_wait_asynccnt(0);
#else
    asm volatile("s_wait_asynccnt 0x0" ::: "memory");
#endif
}

__global__ __launch_bounds__(THREADS)
void dv_bundle_fused(const float* __restrict__ w,
                     const float* __restrict__ x,
                     const float* __restrict__ v,
                     const float* __restrict__ r,
                     float* __restrict__ out)
{
    __shared__ __align__(16) float wlds[2][CH][ROWPITCH];
    __shared__ __align__(16) float xlds[2][CH][4];
    __shared__ float Ipart[WAVES][ROWS][4];
    __shared__ __align__(16) float I4s[ROWS][4];
    __shared__ float Rs[ROWS];

    const int tid  = threadIdx.x;
    const int lane = tid & 31;
    const int wave = tid >> 5;
    const int nb   = blockIdx.x * ROWS;

    const float4* __restrict__ wf4 = (const float4*)w;
    const float4* __restrict__ xf4 = (const float4*)x;

    // WMMA operand swizzle constants (wave32, 16x16x4 f32 layout):
    // A (16x4): lanes 0-15: M=lane, K=0/1 in vgpr0/1; lanes 16-31: K=2/3.
    // B (4x16): vgpr0 = row K0 (lanes 0-15) / K2 (lanes 16-31); vgpr1 = K1 / K3.
    const int  khalf = lane >> 4;
    const int  col3  = lane & 3;
    const int  arow  = lane & 15;
    const bool sel0  = (col3 == khalf * 2);
    const bool sel1  = (col3 == khalf * 2 + 1);

    // Staging / pass-B mapping: 2 rows per wave, 16 consecutive float4 lanes.
    const int rr = tid >> 4;   // 0..15
    const int ii = tid & 15;   // 0..15

    // Issue one chunk's async global->LDS copies (w: 8 x b128/thread, x: 1 x b128/pair).
    auto issue_chunk = [&](int c, int buf) {
        const int i0 = c * CH;
        uint64_t g = (uint64_t)(uintptr_t)(wf4 + (size_t)(nb + rr) * N_INPUTS + (i0 + ii));
        char* l = (char*)&wlds[buf][ii][rr * 4];
        #pragma unroll
        for (int p = 0; p < CH / 16; ++p) {
            async_ld_b128(l, (const void*)(uintptr_t)g);
            g += 16u * 16u;                  // +16 float4 along i
            l += 16 * ROWPITCH * 4;          // +16 LDS i-slots
        }
        if ((tid & 1) == 0) {
            const int xi = tid >> 1;         // 0..127
            async_ld_b128(&xlds[buf][xi][0], xf4 + (i0 + xi));
        }
    };

    v8f acc = {0.f, 0.f, 0.f, 0.f, 0.f, 0.f, 0.f, 0.f};

    // -------- Pass A: I[n,k] = sum_i w[n,i,k]*x[i,k] via WMMA, async double-buffered ----
    issue_chunk(0, 0);
    for (int c = 0; c < NCHUNK; ++c) {
        const int cur = c & 1;
        wait_async0();                       // my chunk-c DMAs landed
        __syncthreads();                     // everyone's landed; prev buffer free
        if (c + 1 < NCHUNK) issue_chunk(c + 1, cur ^ 1);   // overlap DMA with compute

        #pragma unroll
        for (int j = 0; j < IPW; ++j) {
            const int i = wave * IPW + j;
            v2f a  = *(const v2f*)&wlds[cur][i][arow * 4 + khalf * 2];
            v2f xv = *(const v2f*)&xlds[cur][i][khalf * 2];
            v2f b;
            b.x = sel0 ? xv.x : 0.0f;
            b.y = sel1 ? xv.y : 0.0f;
            // D[m,col] += sum_{k} w[nb+m, i, k] * x[i,k] * (col&3==k)
            acc = __builtin_amdgcn_wmma_f32_16x16x4_f32(
                    false, a, false, b, (short)0, acc, false, false);
        }
        __syncthreads();                     // buf[cur] consumed before re-issue at c+2
    }

    // Extract I: element (m, col=k) lives in lane {k | 16+k}, vgpr m&7.
    if ((lane & 15) < 4) {
        #pragma unroll
        for (int j = 0; j < 8; ++j) {
            Ipart[wave][j + 8 * khalf][col3] = acc[j];
        }
    }
    __syncthreads();

    if (tid < ROWS * 4) {
        const int m = tid >> 2, k = tid & 3;
        float s = 0.0f;
        #pragma unroll
        for (int wv = 0; wv < WAVES; ++wv) s += Ipart[wv][m][k];
        I4s[m][k] = s;
    }
    __syncthreads();

    // Per-row scalar dynamics (pre-update v, matching reference ordering).
    if (tid < ROWS) {
        const int n = nb + tid;
        const float I0 = I4s[tid][0], I1 = I4s[tid][1];
        const float I2 = I4s[tid][2], I3 = I4s[tid][3];
        const float vn = v[n], rn = r[n];
        const float dvv  = (I0 - I1 + I2 - I3 - vn) * 0.05f;       // /TAU_V * DT
        const float t    = tanhf(vn);
        const float act  = (vn > 0.0f) ? t : 0.0f;                 // r_new
        const float actd = (vn > 0.0f) ? (1.0f - t * t) : 0.0f;    // ACT_ALPHA=1
        out[n]             = vn + dvv;                             // v_new
        out[N_NEURONS + n] = act;                                  // r_new
        Rs[tid] = rn * actd * dvv * 0.02f;                         // reg / TAU_W
    }
    __syncthreads();

    // -------- Pass B: w_new = w + R*(0.5*x - w*I), streamed (NT read/write) --------
    const float  Rm = Rs[rr];
    const v4f    Im = *(const v4f*)&I4s[rr][0];
    const v4f* __restrict__ wrow = (const v4f*)(wf4 + (size_t)(nb + rr) * N_INPUTS);
    v4f* __restrict__ orow =
        ((v4f*)(out + 2 * N_NEURONS)) + (size_t)(nb + rr) * N_INPUTS;
    const v4f* __restrict__ xrow = (const v4f*)xf4;

    for (int i0 = 0; i0 < N_INPUTS; i0 += 16) {
        const int i = i0 + ii;
        __builtin_prefetch(&wrow[i + 64], 0, 0);             // global_prefetch_b8
        const v4f w4 = __builtin_nontemporal_load(&wrow[i]); // last-use read of w
        const v4f x4 = xrow[i];                              // RT: shared across blocks
        v4f o;
        o.x = fmaf(Rm, fmaf(-w4.x, Im.x, 0.5f * x4.x), w4.x);
        o.y = fmaf(Rm, fmaf(-w4.y, Im.y, 0.5f * x4.y), w4.y);
        o.z = fmaf(Rm, fmaf(-w4.z, Im.z, 0.5f * x4.z), w4.z);
        o.w = fmaf(Rm, fmaf(-w4.w, Im.w, 0.5f * x4.w), w4.w);
        __builtin_nontemporal_store(o, &orow[i]);            // write-once stream
    }
}

extern "C" void kernel_launch(void* const* d_in, const int* in_sizes, int n_in,
                              void* d_out, int out_size, void* d_ws, size_t ws_size,
                              hipStream_t stream)
{
    (void)in_sizes; (void)n_in; (void)out_size; (void)d_ws; (void)ws_size;
    const float* w = (const float*)d_in[0];
    const float* x = (const float*)d_in[1];
    const float* v = (const float*)d_in[2];
    const float* r = (const float*)d_in[3];
    float* out = (float*)d_out;

    dim3 grid(N_NEURONS / ROWS);   // 256 blocks, each owns 16 neuron rows
    dim3 block(THREADS);
    dv_bundle_fused<<<grid, block, 0, stream>>>(w, x, v, r, out);
}